// GNN_basic_26663156973653
// MI455X (gfx1250) — compile-verified
//
#include <hip/hip_runtime.h>
#include <hip/hip_bf16.h>

typedef float v2f __attribute__((ext_vector_type(2)));
typedef float v8f __attribute__((ext_vector_type(8)));

#define GAT_N     50000
#define GAT_E     800000
#define GAT_IN    128
#define GAT_HID   128
#define GAT_HEADS 3
#define GAT_OUT   40
#define GAT_EPS   1e-5f
#define GAT_SLOPE 0.2f

// ---------------------------------------------------------------------------
// order-preserving float<->uint map for atomicMax-based segment max
// ---------------------------------------------------------------------------
__device__ __forceinline__ unsigned enc_f32(float f) {
  unsigned u = __float_as_uint(f);
  return (u & 0x80000000u) ? ~u : (u | 0x80000000u);
}
__device__ __forceinline__ float dec_f32(unsigned e) {
  return (e & 0x80000000u) ? __uint_as_float(e ^ 0x80000000u)
                           : __uint_as_float(~e);
}
__device__ __forceinline__ float leaky(float v) {
  return v > 0.0f ? v : GAT_SLOPE * v;
}

// ---------------------------------------------------------------------------
// GEMM: C[M,Nf] = A[M,K] @ B[K,Nf], fp32 via V_WMMA_F32_16X16X4_F32.
// One wave computes a 16x64 tile (4 accumulators), reusing the A fragment
// across 4 WMMAs per K-step. M % 16 == 0, K % 4 == 0. Out-of-range columns
// load a clamped (safe) address; WMMA only mixes lanes within a column, so
// garbage columns never leak into stored columns. Stores are guarded.
// ---------------------------------------------------------------------------
__global__ __launch_bounds__(32)
void gemm_wmma_f32(const float* __restrict__ A, const float* __restrict__ B,
                   float* __restrict__ C, int M, int K, int Nf) {
  const int lane = threadIdx.x & 31;
  const int r16  = lane & 15;     // row-within-tile for A, col for B/C
  const int khi  = lane >> 4;     // 0: K+0/1, 1: K+2/3 (A/B frag halves)
  const int m0   = blockIdx.x << 4;
  const int n0   = blockIdx.y << 6;   // 64 columns per wave

  int  col[4], colc[4];
  bool ok[4];
#pragma unroll
  for (int j = 0; j < 4; ++j) {
    col[j]  = n0 + 16 * j + r16;
    ok[j]   = (col[j] < Nf);
    colc[j] = ok[j] ? col[j] : 0;     // safe load address
  }

  const float* Arow = A + (size_t)(m0 + r16) * K + (khi << 1);
  const float* Bp   = B + (size_t)(khi << 1) * Nf;

  v8f c[4];
#pragma unroll
  for (int j = 0; j < 4; ++j) c[j] = (v8f){};

  for (int k = 0; k < K; k += 4) {
    v2f a;
    a.x = Arow[k];
    a.y = Arow[k + 1];
    const float* Bk = Bp + (size_t)k * Nf;
#pragma unroll
    for (int j = 0; j < 4; ++j) {
      v2f b;
      b.x = Bk[colc[j]];
      b.y = Bk[Nf + colc[j]];
      c[j] = __builtin_amdgcn_wmma_f32_16x16x4_f32(
          /*neg_a=*/false, a, /*neg_b=*/false, b,
          /*c_mod=*/(short)0, c[j], /*reuse_a=*/false, /*reuse_b=*/false);
    }
  }

  const int rbase = m0 + (khi << 3);   // lanes 16-31 hold rows M=8..15
#pragma unroll
  for (int j = 0; j < 4; ++j) {
    if (ok[j]) {
#pragma unroll
      for (int i = 0; i < 8; ++i)
        C[(size_t)(rbase + i) * Nf + col[j]] = c[j][i];
    }
  }
}

// ---------------------------------------------------------------------------
// el[n,h] = sum_d f[n,h,d]*al[h,d];  er likewise.  One block per node.
// blockDim must be a power of two >= D (<=128).
// ---------------------------------------------------------------------------
__global__ void attn_scores(const float* __restrict__ f,
                            const float* __restrict__ al,
                            const float* __restrict__ ar,
                            float* __restrict__ el, float* __restrict__ er,
                            int H, int D) {
  __shared__ float sd[128];
  const int n = blockIdx.x;
  const int t = threadIdx.x;
  const int bd = blockDim.x;
  for (int h = 0; h < H; ++h) {
    const float v = (t < D) ? f[(size_t)n * H * D + h * D + t] : 0.0f;
    // dot with al
    sd[t] = (t < D) ? v * al[h * D + t] : 0.0f;
    __syncthreads();
    for (int s = bd >> 1; s > 0; s >>= 1) {
      if (t < s) sd[t] += sd[t + s];
      __syncthreads();
    }
    if (t == 0) el[(size_t)n * H + h] = sd[0];
    __syncthreads();
    // dot with ar
    sd[t] = (t < D) ? v * ar[h * D + t] : 0.0f;
    __syncthreads();
    for (int s = bd >> 1; s > 0; s >>= 1) {
      if (t < s) sd[t] += sd[t + s];
      __syncthreads();
    }
    if (t == 0) er[(size_t)n * H + h] = sd[0];
    __syncthreads();
  }
}

// init m_enc=0 (== below enc(-inf)) and s=0
__global__ void init_ms(unsigned* __restrict__ m_enc, float* __restrict__ s,
                        int count) {
  int i = blockIdx.x * blockDim.x + threadIdx.x;
  if (i < count) { m_enc[i] = 0u; s[i] = 0.0f; }
}

// pass 1: segment max of leaky(el[src]+er[dst]) over dst
__global__ void edge_max(const int* __restrict__ src, const int* __restrict__ dst,
                         const float* __restrict__ el, const float* __restrict__ er,
                         unsigned* __restrict__ m_enc, int E, int H) {
  int i = blockIdx.x * blockDim.x + threadIdx.x;
  if (i >= E * H) return;
  const int e = i / H, h = i - e * H;
  const int sN = src[e], dN = dst[e];
  const float v = leaky(el[(size_t)sN * H + h] + er[(size_t)dN * H + h]);
  atomicMax(&m_enc[(size_t)dN * H + h], enc_f32(v));
}

// decode max in-place: uint encoding -> float
__global__ void finalize_max(unsigned* __restrict__ m_enc, int count) {
  int i = blockIdx.x * blockDim.x + threadIdx.x;
  if (i < count) {
    float v = dec_f32(m_enc[i]);
    ((float*)m_enc)[i] = v;
  }
}

// pass 2: ex = exp(e - m[dst]); s[dst] += ex
__global__ void edge_exp(const int* __restrict__ src, const int* __restrict__ dst,
                         const float* __restrict__ el, const float* __restrict__ er,
                         const float* __restrict__ m, float* __restrict__ ex,
                         float* __restrict__ s, int E, int H) {
  int i = blockIdx.x * blockDim.x + threadIdx.x;
  if (i >= E * H) return;
  const int e = i / H, h = i - e * H;
  const int sN = src[e], dN = dst[e];
  const float v  = leaky(el[(size_t)sN * H + h] + er[(size_t)dN * H + h]);
  const float xv = expf(v - m[(size_t)dN * H + h]);
  ex[i] = xv;
  atomicAdd(&s[(size_t)dN * H + h], xv);
}

// out[n, c] = bias[c]
__global__ void init_out(float* __restrict__ out, const float* __restrict__ bias,
                         int count, int F) {
  int i = blockIdx.x * blockDim.x + threadIdx.x;
  if (i < count) out[i] = bias[i % F];
}

// pass 3: out[dst] += f[src] * (ex / s[dst]) per head/feature. Block per edge.
__global__ void edge_aggregate(const int* __restrict__ src, const int* __restrict__ dst,
                               const float* __restrict__ f, const float* __restrict__ ex,
                               const float* __restrict__ s, float* __restrict__ out,
                               int H, int D) {
  const int e = blockIdx.x;
  const int t = threadIdx.x;
  if (t >= D) return;
  const int sN = src[e], dN = dst[e];
  for (int h = 0; h < H; ++h) {
    const float a = ex[(size_t)e * H + h] / s[(size_t)dN * H + h];
    atomicAdd(&out[(size_t)dN * H * D + h * D + t],
              f[(size_t)sN * H * D + h * D + t] * a);
  }
}

// per-channel mean / inv-std; one block (256 thr) per channel
__global__ __launch_bounds__(256)
void bn_stats(const float* __restrict__ h, float* __restrict__ mean,
              float* __restrict__ inv, int Nn, int C) {
  __shared__ float ssum[256];
  __shared__ float ssq[256];
  const int c = blockIdx.x;
  const int t = threadIdx.x;
  float sum = 0.0f, sq = 0.0f;
  for (int r = t; r < Nn; r += 256) {
    const float v = h[(size_t)r * C + c];
    sum += v; sq += v * v;
  }
  ssum[t] = sum; ssq[t] = sq;
  __syncthreads();
  for (int s = 128; s > 0; s >>= 1) {
    if (t < s) { ssum[t] += ssum[t + s]; ssq[t] += ssq[t + s]; }
    __syncthreads();
  }
  if (t == 0) {
    const float mu  = ssum[0] / (float)Nn;
    const float var = ssq[0] / (float)Nn - mu * mu;
    mean[c] = mu;
    inv[c]  = rsqrtf(var + GAT_EPS);
  }
}

// h = relu(g * (h - mean) * inv + be), in place
__global__ void bn_relu(float* __restrict__ h, const float* __restrict__ mean,
                        const float* __restrict__ inv, const float* __restrict__ g,
                        const float* __restrict__ be, int count, int C) {
  int i = blockIdx.x * blockDim.x + threadIdx.x;
  if (i >= count) return;
  const int c = i % C;
  const float v = g[c] * (h[i] - mean[c]) * inv[c] + be[c];
  h[i] = v > 0.0f ? v : 0.0f;
}

// ---------------------------------------------------------------------------
// orchestration
// ---------------------------------------------------------------------------
static inline int cdiv(int a, int b) { return (a + b - 1) / b; }

extern "C" void kernel_launch(void* const* d_in, const int* in_sizes, int n_in,
                              void* d_out, int out_size, void* d_ws, size_t ws_size,
                              hipStream_t stream) {
  (void)in_sizes; (void)n_in; (void)out_size; (void)ws_size;

  const float* x   = (const float*)d_in[0];
  const int*   src = (const int*)  d_in[1];
  const int*   dst = (const int*)  d_in[2];
  const float* W0  = (const float*)d_in[3];
  const float* al0 = (const float*)d_in[4];
  const float* ar0 = (const float*)d_in[5];
  const float* b0  = (const float*)d_in[6];
  const float* W1  = (const float*)d_in[7];
  const float* al1 = (const float*)d_in[8];
  const float* ar1 = (const float*)d_in[9];
  const float* b1  = (const float*)d_in[10];
  const float* W2  = (const float*)d_in[11];
  const float* al2 = (const float*)d_in[12];
  const float* ar2 = (const float*)d_in[13];
  const float* b2  = (const float*)d_in[14];
  const float* g0  = (const float*)d_in[15];
  const float* be0 = (const float*)d_in[16];
  const float* g1  = (const float*)d_in[17];
  const float* be1 = (const float*)d_in[18];

  const int N = GAT_N, E = GAT_E, H = GAT_HEADS;
  const int F = GAT_HEADS * GAT_HID;   // 384

  // workspace layout (floats)
  float* ws   = (float*)d_ws;
  float* bufF = ws;                         // N*F   : f = h @ W
  float* bufO = bufF + (size_t)N * F;       // N*F   : aggregated output / h
  float* el   = bufO + (size_t)N * F;       // N*H
  float* er   = el   + (size_t)N * H;       // N*H
  float* m    = er   + (size_t)N * H;       // N*H   (uint during max phase)
  float* s    = m    + (size_t)N * H;       // N*H
  float* ex   = s    + (size_t)N * H;       // E*H
  float* mean = ex   + (size_t)E * H;       // F
  float* inv  = mean + F;                   // F

  const dim3 blk256(256);

  // ---------------- layer 0: IN=128 -> H*HID=384 ----------------
  gemm_wmma_f32<<<dim3(N / 16, cdiv(F, 64)), dim3(32), 0, stream>>>(x, W0, bufF, N, GAT_IN, F);
  attn_scores<<<dim3(N), dim3(128), 0, stream>>>(bufF, al0, ar0, el, er, H, GAT_HID);
  init_ms<<<dim3(cdiv(N * H, 256)), blk256, 0, stream>>>((unsigned*)m, s, N * H);
  edge_max<<<dim3(cdiv(E * H, 256)), blk256, 0, stream>>>(src, dst, el, er, (unsigned*)m, E, H);
  finalize_max<<<dim3(cdiv(N * H, 256)), blk256, 0, stream>>>((unsigned*)m, N * H);
  edge_exp<<<dim3(cdiv(E * H, 256)), blk256, 0, stream>>>(src, dst, el, er, m, ex, s, E, H);
  init_out<<<dim3(cdiv(N * F, 256)), blk256, 0, stream>>>(bufO, b0, N * F, F);
  edge_aggregate<<<dim3(E), dim3(128), 0, stream>>>(src, dst, bufF, ex, s, bufO, H, GAT_HID);
  bn_stats<<<dim3(F), blk256, 0, stream>>>(bufO, mean, inv, N, F);
  bn_relu<<<dim3(cdiv(N * F, 256)), blk256, 0, stream>>>(bufO, mean, inv, g0, be0, N * F, F);

  // ---------------- layer 1: 384 -> 384 ----------------
  gemm_wmma_f32<<<dim3(N / 16, cdiv(F, 64)), dim3(32), 0, stream>>>(bufO, W1, bufF, N, F, F);
  attn_scores<<<dim3(N), dim3(128), 0, stream>>>(bufF, al1, ar1, el, er, H, GAT_HID);
  init_ms<<<dim3(cdiv(N * H, 256)), blk256, 0, stream>>>((unsigned*)m, s, N * H);
  edge_max<<<dim3(cdiv(E * H, 256)), blk256, 0, stream>>>(src, dst, el, er, (unsigned*)m, E, H);
  finalize_max<<<dim3(cdiv(N * H, 256)), blk256, 0, stream>>>((unsigned*)m, N * H);
  edge_exp<<<dim3(cdiv(E * H, 256)), blk256, 0, stream>>>(src, dst, el, er, m, ex, s, E, H);
  init_out<<<dim3(cdiv(N * F, 256)), blk256, 0, stream>>>(bufO, b1, N * F, F);
  edge_aggregate<<<dim3(E), dim3(128), 0, stream>>>(src, dst, bufF, ex, s, bufO, H, GAT_HID);
  bn_stats<<<dim3(F), blk256, 0, stream>>>(bufO, mean, inv, N, F);
  bn_relu<<<dim3(cdiv(N * F, 256)), blk256, 0, stream>>>(bufO, mean, inv, g1, be1, N * F, F);

  // ---------------- layer 2: 384 -> 40 (1 head), out -> d_out ----------------
  float* out = (float*)d_out;
  gemm_wmma_f32<<<dim3(N / 16, cdiv(GAT_OUT, 64)), dim3(32), 0, stream>>>(bufO, W2, bufF, N, F, GAT_OUT);
  attn_scores<<<dim3(N), dim3(64), 0, stream>>>(bufF, al2, ar2, el, er, 1, GAT_OUT);
  init_ms<<<dim3(cdiv(N, 256)), blk256, 0, stream>>>((unsigned*)m, s, N);
  edge_max<<<dim3(cdiv(E, 256)), blk256, 0, stream>>>(src, dst, el, er, (unsigned*)m, E, 1);
  finalize_max<<<dim3(cdiv(N, 256)), blk256, 0, stream>>>((unsigned*)m, N);
  edge_exp<<<dim3(cdiv(E, 256)), blk256, 0, stream>>>(src, dst, el, er, m, ex, s, E, 1);
  init_out<<<dim3(cdiv(N * GAT_OUT, 256)), blk256, 0, stream>>>(out, b2, N * GAT_OUT, GAT_OUT);
  edge_aggregate<<<dim3(E), dim3(64), 0, stream>>>(src, dst, bufF, ex, s, out, 1, GAT_OUT);
}